// MultiHeadMinkUnet_87814901334087
// MI455X (gfx1250) — compile-verified
//
#include <hip/hip_runtime.h>
#include <hip/hip_bf16.h>

// ---------------------------------------------------------------------------
// Problem constants (match reference)
// ---------------------------------------------------------------------------
#define NPTS   1000000
#define BATCH  4
#define SPG    1024          // superpoints per batch
#define GRP    (BATCH * SPG) // 4096 groups
#define DMODEL 96
#define NHEAD  4
#define DH     24            // DMODEL / NHEAD
#define NLAB   20
#define NUNLAB 30
#define NOUTC  (DMODEL + NLAB + NUNLAB) // 146

typedef __attribute__((ext_vector_type(2))) float v2f;
typedef __attribute__((ext_vector_type(8))) float v8f;

// V_WMMA_F32_16X16X4_F32 : D = A(16x4) * B(4x16) + C(16x16), fp32 throughout.
__device__ __forceinline__ v8f wmma4(v2f a, v2f b, v8f c) {
    return __builtin_amdgcn_wmma_f32_16x16x4_f32(false, a, false, b, (short)0, c,
                                                 false, false);
}
__device__ __forceinline__ v2f mkv2(float x, float y) {
    v2f r; r.x = x; r.y = y; return r;
}

// ---------------------------------------------------------------------------
// 1) zero counts + Tsum regions of workspace
// ---------------------------------------------------------------------------
__global__ __launch_bounds__(256) void k_zero(float* p, int n) {
    int i = blockIdx.x * 256 + threadIdx.x;
    if (i < n) p[i] = 0.0f;
}

// ---------------------------------------------------------------------------
// 2) segment sum: counts[g] += 1, Tsum[g][:] += feats[i][:]
//    8 points per 256-thread block, one wave per point, 3 atomics/lane.
//    Tsum table is 1.5MB -> all atomics resolve in L2.
// ---------------------------------------------------------------------------
__global__ __launch_bounds__(256) void k_accum(const float* __restrict__ feats,
                                               const int* __restrict__ bi,
                                               const int* __restrict__ spi,
                                               float* counts, float* Tsum) {
    int wave = threadIdx.x >> 5;
    int lane = threadIdx.x & 31;
    int i = blockIdx.x * 8 + wave;
    if (i >= NPTS) return;
    int g = bi[i] * SPG + spi[i];
    const float* src = feats + (size_t)i * DMODEL;
    float* dst = Tsum + (size_t)g * DMODEL;
    atomicAdd(&dst[lane], src[lane]);
    atomicAdd(&dst[lane + 32], src[lane + 32]);
    atomicAdd(&dst[lane + 64], src[lane + 64]);
    if (lane == 0) atomicAdd(&counts[g], 1.0f);
}

// ---------------------------------------------------------------------------
// 3) T = Tsum / max(counts, 1)   (in place)
// ---------------------------------------------------------------------------
__global__ __launch_bounds__(256) void k_finalize(float* T, const float* counts) {
    int i = blockIdx.x * 256 + threadIdx.x;
    if (i < GRP * DMODEL) {
        float c = counts[i / DMODEL];
        T[i] = T[i] / fmaxf(c, 1.0f);
    }
}

// ---------------------------------------------------------------------------
// 4)+(6) generic wave-level GEMM tile: O = (Cin?) + A[4096,96] @ W^T
//    grid = (256 row-tiles, 6 col-tiles, nz);  z selects among (W,O) triples.
//    One wave per 16x16 tile.  Permuted-K scheme: each lane loads one b128
//    (cols 8kc + half*4 ..+3) which feeds TWO wmma chunks -> 24 WMMA from
//    24 b128 loads (K=96).
// ---------------------------------------------------------------------------
__global__ __launch_bounds__(32) void k_gemm_tw(const float* __restrict__ A,
                                                const float* __restrict__ W0,
                                                const float* __restrict__ W1,
                                                const float* __restrict__ W2,
                                                const float* __restrict__ Cin,
                                                float* O0, float* O1, float* O2) {
    int lane = threadIdx.x & 31;
    int half = lane >> 4;
    int lr   = lane & 15;
    int mBase = blockIdx.x * 16;
    int nBase = blockIdx.y * 16;
    const float* W = (blockIdx.z == 0) ? W0 : (blockIdx.z == 1) ? W1 : W2;
    float* O       = (blockIdx.z == 0) ? O0 : (blockIdx.z == 1) ? O1 : O2;

    v8f acc;
    for (int r = 0; r < 8; ++r) {
        int m = r + half * 8;
        acc[r] = Cin ? Cin[(size_t)(mBase + m) * DMODEL + nBase + lr] : 0.0f;
    }
    const float* Arow = A + (size_t)(mBase + lr) * DMODEL;
    const float* Wrow = W + (size_t)(nBase + lr) * DMODEL;
    #pragma unroll
    for (int kc = 0; kc < 12; ++kc) {
        int k0 = kc * 8 + half * 4;
        float4 aa = *(const float4*)(Arow + k0);
        float4 bb = *(const float4*)(Wrow + k0);
        acc = wmma4(mkv2(aa.x, aa.y), mkv2(bb.x, bb.y), acc);
        acc = wmma4(mkv2(aa.z, aa.w), mkv2(bb.z, bb.w), acc);
    }
    for (int r = 0; r < 8; ++r) {
        int m = r + half * 8;
        O[(size_t)(mBase + m) * DMODEL + nBase + lr] = acc[r];
    }
}

// ---------------------------------------------------------------------------
// 5) attention: one wave per (batch, head, 16-query tile), streaming 1024 keys.
//    Flash-style online softmax.  QK^T uses permuted-K b128 loads of K rows.
//    V tiles (16x24) are staged to LDS with GLOBAL_LOAD_ASYNC_TO_LDS_B128
//    (3 wave-instructions per tile) + s_wait_asynccnt, then P@V reads B
//    operands from LDS (conflict-free: 24-float row stride).
// ---------------------------------------------------------------------------
__global__ __launch_bounds__(32) void k_attn(const float* __restrict__ Qm,
                                             const float* __restrict__ Km,
                                             const float* __restrict__ Vm,
                                             float* __restrict__ Oh) {
    __shared__ float pl[16][17];
    __shared__ __align__(16) float vlds[16][24];
    int lane = threadIdx.x & 31;
    int half = lane >> 4;
    int lr   = lane & 15;
    int mt = blockIdx.x, h = blockIdx.y, b = blockIdx.z;
    int rowBase = b * SPG + mt * 16;
    const float SCALE = 0.2041241452319315f; // 1/sqrt(24)

    // async-copy lane mapping for the 16x24 V tile: 96 float4 chunks, 3 waves
    int vrow[3], vchk[3];
    unsigned vldsoff[3];
    #pragma unroll
    for (int t = 0; t < 3; ++t) {
        int idx = t * 32 + lane;
        vrow[t] = idx / 6;
        vchk[t] = idx - vrow[t] * 6;
        vldsoff[t] = (unsigned)(unsigned long long)(&vlds[vrow[t]][vchk[t] * 4]);
    }

    // Q tile [16 x 24] as 3 b128 per lane, pre-scaled
    float4 q4[3];
    {
        const float* qrow = Qm + (size_t)(rowBase + lr) * DMODEL + h * DH;
        #pragma unroll
        for (int t = 0; t < 3; ++t) {
            float4 q = *(const float4*)(qrow + t * 8 + half * 4);
            q4[t].x = q.x * SCALE; q4[t].y = q.y * SCALE;
            q4[t].z = q.z * SCALE; q4[t].w = q.w * SCALE;
        }
    }

    float oldmax[8], denom[8];
    v8f o0, o1;
    for (int r = 0; r < 8; ++r) { oldmax[r] = -3.0e38f; denom[r] = 0.0f; o0[r] = 0.0f; o1[r] = 0.0f; }

    for (int kt = 0; kt < SPG / 16; ++kt) {
        int kbase = b * SPG + kt * 16;

        // make sure last tile's LDS reads finished, then kick off async V copy
        asm volatile("s_wait_dscnt 0x0" ::: "memory");
        #pragma unroll
        for (int t = 0; t < 3; ++t) {
            const float* gp = Vm + (size_t)(kbase + vrow[t]) * DMODEL + h * DH + vchk[t] * 4;
            asm volatile("global_load_async_to_lds_b128 %0, %1, off"
                         :: "v"(vldsoff[t]), "v"(gp) : "memory");
        }

        // S tile = Q @ K^T  (6 WMMA from 3 b128 K-row loads per lane)
        v8f s;
        for (int r = 0; r < 8; ++r) s[r] = 0.0f;
        {
            const float* krow = Km + (size_t)(kbase + lr) * DMODEL + h * DH;
            #pragma unroll
            for (int t = 0; t < 3; ++t) {
                float4 kk = *(const float4*)(krow + t * 8 + half * 4);
                s = wmma4(mkv2(q4[t].x, q4[t].y), mkv2(kk.x, kk.y), s);
                s = wmma4(mkv2(q4[t].z, q4[t].w), mkv2(kk.z, kk.w), s);
            }
        }
        // online softmax (row = r + 8*half, reductions across 16 lanes of half)
        #pragma unroll
        for (int r = 0; r < 8; ++r) {
            float sv = s[r];
            float m = sv;
            m = fmaxf(m, __shfl_xor(m, 1));
            m = fmaxf(m, __shfl_xor(m, 2));
            m = fmaxf(m, __shfl_xor(m, 4));
            m = fmaxf(m, __shfl_xor(m, 8));
            float nm = fmaxf(oldmax[r], m);
            float alpha = __expf(oldmax[r] - nm);
            float p = __expf(sv - nm);
            float rs = p;
            rs += __shfl_xor(rs, 1);
            rs += __shfl_xor(rs, 2);
            rs += __shfl_xor(rs, 4);
            rs += __shfl_xor(rs, 8);
            denom[r] = denom[r] * alpha + rs;
            oldmax[r] = nm;
            o0[r] *= alpha;
            o1[r] *= alpha;
            pl[r + half * 8][lr] = p;   // D-layout -> LDS
        }
        __syncthreads();
        v2f pa[4];                       // LDS -> A-layout
        #pragma unroll
        for (int kc = 0; kc < 4; ++kc) {
            int k0 = kc * 4 + half * 2;
            pa[kc].x = pl[lr][k0];
            pa[kc].y = pl[lr][k0 + 1];
        }
        __syncthreads();

        // V tile must be resident in LDS before P@V
        asm volatile("s_wait_asynccnt 0x0" ::: "memory");

        // O += P @ V : 2 N-tiles (24 cols padded to 32), 4 k-chunks each
        #pragma unroll
        for (int j = 0; j < 2; ++j) {
            #pragma unroll
            for (int kc = 0; kc < 4; ++kc) {
                int kk = kc * 4 + half * 2;
                int n = j * 16 + lr;
                float bx = 0.0f, by = 0.0f;
                if (n < DH) {
                    bx = vlds[kk][n];
                    by = vlds[kk + 1][n];
                }
                v2f bv = mkv2(bx, by);
                if (j == 0) o0 = wmma4(pa[kc], bv, o0);
                else        o1 = wmma4(pa[kc], bv, o1);
            }
        }
    }
    // normalize and emit O head slice [16 x 24]
    for (int r = 0; r < 8; ++r) {
        float inv = 1.0f / denom[r];
        int row = rowBase + r + half * 8;
        Oh[(size_t)row * DMODEL + h * DH + lr] = o0[r] * inv;
        if (lr < 8)
            Oh[(size_t)row * DMODEL + h * DH + 16 + lr] = o1[r] * inv;
    }
}

// ---------------------------------------------------------------------------
// 7) streaming output pass: out[:, :96] = feats + Z[g]; out[:, 96:146] = heads.
//    4 waves per 128-thread block, one 16-row tile per wave.  Permuted-K b128
//    loads: each (feats,Z) b128 pair feeds the residual store and two WMMA
//    chunks of the 50-wide head GEMM (4 padded 16-col tiles, K=96).
// ---------------------------------------------------------------------------
__global__ __launch_bounds__(128) void k_out(const float* __restrict__ feats,
                                             const int* __restrict__ bi,
                                             const int* __restrict__ spi,
                                             const float* __restrict__ Z,
                                             const float* __restrict__ Wlab,
                                             const float* __restrict__ Wunlab,
                                             float* __restrict__ out) {
    int lane = threadIdx.x & 31;
    int half = lane >> 4;
    int lr   = lane & 15;
    int tile = blockIdx.x * 4 + (threadIdx.x >> 5);
    int mBase = tile * 16;
    int row = mBase + lr;

    int g = bi[row] * SPG + spi[row];
    const float* frow = feats + (size_t)row * DMODEL;
    const float* zrow = Z + (size_t)g * DMODEL;
    float* orow = out + (size_t)row * NOUTC;

    v8f acc[4];
    for (int t = 0; t < 4; ++t)
        for (int r = 0; r < 8; ++r) acc[t][r] = 0.0f;

    #pragma unroll
    for (int kc = 0; kc < 12; ++kc) {
        int k0 = kc * 8 + half * 4;
        float4 f = *(const float4*)(frow + k0);
        float4 z = *(const float4*)(zrow + k0);
        float4 a4;
        a4.x = f.x + z.x; a4.y = f.y + z.y;
        a4.z = f.z + z.z; a4.w = f.w + z.w;
        // residual columns (row stride 584B is 8B-aligned -> 2x b64 stores)
        *(float2*)(orow + k0)     = make_float2(a4.x, a4.y);
        *(float2*)(orow + k0 + 2) = make_float2(a4.z, a4.w);
        v2f a0 = mkv2(a4.x, a4.y);
        v2f a1 = mkv2(a4.z, a4.w);
        #pragma unroll
        for (int nt = 0; nt < 4; ++nt) {
            int n = nt * 16 + lr;
            float4 bb;
            bb.x = 0.0f; bb.y = 0.0f; bb.z = 0.0f; bb.w = 0.0f;
            if (n < NLAB) {
                bb = *(const float4*)(Wlab + (size_t)n * DMODEL + k0);
            } else if (n < NLAB + NUNLAB) {
                bb = *(const float4*)(Wunlab + (size_t)(n - NLAB) * DMODEL + k0);
            }
            acc[nt] = wmma4(a0, mkv2(bb.x, bb.y), acc[nt]);
            acc[nt] = wmma4(a1, mkv2(bb.z, bb.w), acc[nt]);
        }
    }
    // logits columns 96..145
    for (int nt = 0; nt < 4; ++nt) {
        int n = nt * 16 + lr;
        if (n < NLAB + NUNLAB) {
            for (int r = 0; r < 8; ++r) {
                int m = mBase + r + half * 8;
                out[(size_t)m * NOUTC + DMODEL + n] = acc[nt][r];
            }
        }
    }
}

// ---------------------------------------------------------------------------
// launcher
// ---------------------------------------------------------------------------
extern "C" void kernel_launch(void* const* d_in, const int* in_sizes, int n_in,
                              void* d_out, int out_size, void* d_ws, size_t ws_size,
                              hipStream_t stream) {
    const float* feats  = (const float*)d_in[0];
    // d_in[1] = xyz : only feeds dead code (dist/R) in the reference -> skipped
    const int*   bi     = (const int*)d_in[2];
    const int*   spi    = (const int*)d_in[3];
    const float* Wq     = (const float*)d_in[4];
    const float* Wk     = (const float*)d_in[5];
    const float* Wv     = (const float*)d_in[6];
    const float* Wo     = (const float*)d_in[7];
    const float* Wlab   = (const float*)d_in[8];
    const float* Wunlab = (const float*)d_in[9];
    float* out = (float*)d_out;

    float* wsf    = (float*)d_ws;
    float* counts = wsf;                         // GRP
    float* T      = counts + GRP;                // GRP*DMODEL  (Tsum -> T in place)
    float* Qm     = T  + (size_t)GRP * DMODEL;
    float* Km     = Qm + (size_t)GRP * DMODEL;
    float* Vm     = Km + (size_t)GRP * DMODEL;
    float* Oh     = Vm + (size_t)GRP * DMODEL;
    float* Zm     = Oh + (size_t)GRP * DMODEL;

    // 1) zero counts + Tsum
    {
        int n = GRP + GRP * DMODEL;
        k_zero<<<(n + 255) / 256, 256, 0, stream>>>(counts, n);
    }
    // 2) segment sums (L2 atomics)
    k_accum<<<NPTS / 8, 256, 0, stream>>>(feats, bi, spi, counts, T);
    // 3) means
    k_finalize<<<(GRP * DMODEL + 255) / 256, 256, 0, stream>>>(T, counts);
    // 4) Q,K,V projections (WMMA f32 16x16x4)
    k_gemm_tw<<<dim3(GRP / 16, DMODEL / 16, 3), 32, 0, stream>>>(
        T, Wq, Wk, Wv, nullptr, Qm, Km, Vm);
    // 5) per-(batch,head) flash attention over 1024 superpoints
    k_attn<<<dim3(SPG / 16, NHEAD, BATCH), 32, 0, stream>>>(Qm, Km, Vm, Oh);
    // 6) Z = T + O @ Wo^T
    k_gemm_tw<<<dim3(GRP / 16, DMODEL / 16, 1), 32, 0, stream>>>(
        Oh, Wo, Wo, Wo, T, Zm, Zm, Zm);
    // 7) fused residual + heads, streaming over all 1M points
    k_out<<<NPTS / 64, 128, 0, stream>>>(feats, bi, spi, Zm, Wlab, Wunlab, out);
}